// CNE_Minus_35433480192875
// MI455X (gfx1250) — compile-verified
//
#include <hip/hip_runtime.h>
#include <hip/hip_bf16.h>

typedef __attribute__((ext_vector_type(2))) float v2f;
typedef __attribute__((ext_vector_type(8))) float v8f;

#define HDIM 64
#define INDIM 128

__device__ __forceinline__ float lrelu(float v) { return v > 0.0f ? v : 0.01f * v; }

// ---------------- degree / normalization ----------------
__global__ void deg_init_kernel(float* __restrict__ deg, int n) {
    int i = blockIdx.x * blockDim.x + threadIdx.x;
    if (i < n) deg[i] = 1.0f;  // self-loop contribution
}

__global__ void deg_count_kernel(const int* __restrict__ dst, float* __restrict__ deg, int e) {
    int i = blockIdx.x * blockDim.x + threadIdx.x;
    if (i < e)
        __hip_atomic_fetch_add(&deg[dst[i]], 1.0f, __ATOMIC_RELAXED, __HIP_MEMORY_SCOPE_AGENT);
}

__global__ void dinv_kernel(float* __restrict__ deg, int n) {
    int i = blockIdx.x * blockDim.x + threadIdx.x;
    if (i < n) {
        float d = deg[i];
        deg[i] = d > 0.0f ? rsqrtf(d) : 0.0f;
    }
}

// ---------------- fp32 WMMA GEMM: C[M,64] = A[M,K] @ B[K,64] (+bias)(+lrelu) ----------------
// One wave computes a full 16x64 row strip (4 accumulators). B is staged in LDS in
// WMMA-fragment order so each k-step needs only: 1 global b64 (A) + 2 ds b128 (B) + 4 WMMA.
__global__ void wmma_gemm_f32_kernel(const float* __restrict__ A, const float* __restrict__ B,
                                     float* __restrict__ C, const float* __restrict__ bias,
                                     int M, int K, int act) {
    extern __shared__ float bsw[];  // K*64 floats, swizzled: [ks][lane][tn] pairs
    const int tid   = threadIdx.x;
    const int lane  = tid & 31;
    const int wave  = tid >> 5;
    const int lmod  = lane & 15;
    const int lhalf = lane >> 4;  // 0 -> K+{0,1}, 1 -> K+{2,3}

    // Stage B swizzled: pair p = ((ks*32 + l)*4 + tn); values B[kb..kb+1][tn*16+lmod(l)]
    const int pairs = K * 32;  // multiple of 256 for K in {64,128}
    for (int p = tid; p < pairs; p += 256) {
        int ks  = p >> 7;
        int ll  = (p >> 2) & 31;
        int tn  = p & 3;
        int kb  = 4 * ks + 2 * (ll >> 4);
        int col = tn * 16 + (ll & 15);
        bsw[p * 2 + 0] = B[(size_t)kb * HDIM + col];
        bsw[p * 2 + 1] = B[(size_t)(kb + 1) * HDIM + col];
    }
    __syncthreads();

    const int gwave = blockIdx.x * 8 + wave;
    if (gwave >= (M >> 4)) return;  // wave-uniform: EXEC all-1s at every WMMA

    const float* arow = A + (size_t)(gwave * 16 + lmod) * K + 2 * lhalf;
    v8f acc0 = {}, acc1 = {}, acc2 = {}, acc3 = {};

    const int steps = K >> 2;
    for (int ks = 0; ks < steps; ++ks) {
        v2f a = *(const v2f*)(arow + 4 * ks);                       // global_load_b64, imm offset
        const float4* bp = (const float4*)(bsw + (size_t)(ks * 32 + lane) * 8);
        float4 b01 = bp[0];                                         // ds_load_b128
        float4 b23 = bp[1];                                         // ds_load_b128
        v2f b0 = {b01.x, b01.y}, b1 = {b01.z, b01.w};
        v2f b2 = {b23.x, b23.y}, b3 = {b23.z, b23.w};
        acc0 = __builtin_amdgcn_wmma_f32_16x16x4_f32(false, a, false, b0, (short)0, acc0, false, false);
        acc1 = __builtin_amdgcn_wmma_f32_16x16x4_f32(false, a, false, b1, (short)0, acc1, false, false);
        acc2 = __builtin_amdgcn_wmma_f32_16x16x4_f32(false, a, false, b2, (short)0, acc2, false, false);
        acc3 = __builtin_amdgcn_wmma_f32_16x16x4_f32(false, a, false, b3, (short)0, acc3, false, false);
    }

    // C/D layout: VGPR r -> row r (lanes 0-15) / r+8 (lanes 16-31), col = lane%16 within tile
    const int rbase = gwave * 16 + lhalf * 8;
    float bi0 = 0.f, bi1 = 0.f, bi2 = 0.f, bi3 = 0.f;
    if (bias) { bi0 = bias[lmod]; bi1 = bias[16 + lmod]; bi2 = bias[32 + lmod]; bi3 = bias[48 + lmod]; }
#pragma unroll
    for (int r = 0; r < 8; ++r) {
        size_t ro = (size_t)(rbase + r) * HDIM + lmod;
        float v0 = acc0[r] + bi0, v1 = acc1[r] + bi1, v2 = acc2[r] + bi2, v3 = acc3[r] + bi3;
        if (act) { v0 = lrelu(v0); v1 = lrelu(v1); v2 = lrelu(v2); v3 = lrelu(v3); }
        C[ro] = v0; C[ro + 16] = v1; C[ro + 32] = v2; C[ro + 48] = v3;
    }
}

// ---------------- GCN conv pieces ----------------
// out[i,:] = bias + dinv[i]^2 * H[i,:]   (self-loop message + bias, resets accumulator)
__global__ void conv_init_kernel(const float* __restrict__ Hm, const float* __restrict__ dinv,
                                 const float* __restrict__ bias, float* __restrict__ out, int n) {
    int tid = blockIdx.x * blockDim.x + threadIdx.x;  // n*16 threads, float4 per thread
    if (tid >= n * 16) return;
    int i = tid >> 4, j = (tid & 15) * 4;
    float s = dinv[i]; s *= s;
    float4 h = *(const float4*)(Hm + (size_t)i * HDIM + j);
    float4 b = *(const float4*)(bias + j);
    float4 o = make_float4(b.x + s * h.x, b.y + s * h.y, b.z + s * h.z, b.w + s * h.w);
    *(float4*)(out + (size_t)i * HDIM + j) = o;
}

// out[dst,:] += dinv[src]*dinv[dst] * H[src,:]  — 16 threads/edge, float4 chunks
__global__ void conv_scatter_kernel(const float* __restrict__ Hm, const int* __restrict__ src,
                                    const int* __restrict__ dst, const float* __restrict__ dinv,
                                    float* __restrict__ out, int e) {
    int tid = blockIdx.x * blockDim.x + threadIdx.x;
    int ed = tid >> 4;
    if (ed >= e) return;
    int j = (tid & 15) * 4;
    int s = src[ed], d = dst[ed];
    float nrm = dinv[s] * dinv[d];
    float4 h = *(const float4*)(Hm + (size_t)s * HDIM + j);
    float* o = out + (size_t)d * HDIM + j;
    __hip_atomic_fetch_add(o + 0, nrm * h.x, __ATOMIC_RELAXED, __HIP_MEMORY_SCOPE_AGENT);
    __hip_atomic_fetch_add(o + 1, nrm * h.y, __ATOMIC_RELAXED, __HIP_MEMORY_SCOPE_AGENT);
    __hip_atomic_fetch_add(o + 2, nrm * h.z, __ATOMIC_RELAXED, __HIP_MEMORY_SCOPE_AGENT);
    __hip_atomic_fetch_add(o + 3, nrm * h.w, __ATOMIC_RELAXED, __HIP_MEMORY_SCOPE_AGENT);
}

__global__ void relu_inplace_kernel(float* __restrict__ buf, int n16) {
    int tid = blockIdx.x * blockDim.x + threadIdx.x;
    if (tid >= n16) return;
    float4* p = (float4*)buf + tid;
    float4 v = *p;
    v.x = fmaxf(v.x, 0.0f); v.y = fmaxf(v.y, 0.0f);
    v.z = fmaxf(v.z, 0.0f); v.w = fmaxf(v.w, 0.0f);
    *p = v;
}

// ---------------- outcome heads: two 64-dot products per gathered node ----------------
__global__ void dual_head_kernel(const float* __restrict__ Z, const int* __restrict__ idx, int n,
                                 const float* __restrict__ Wa, const float* __restrict__ ba,
                                 float* __restrict__ outA,
                                 const float* __restrict__ Wb, const float* __restrict__ bb,
                                 float* __restrict__ outB) {
    int t = blockIdx.x * blockDim.x + threadIdx.x;
    if (t >= n) return;
    const float* row = Z + (size_t)idx[t] * HDIM;
    float da = 0.0f, db = 0.0f;
#pragma unroll
    for (int j = 0; j < HDIM; j += 4) {
        float4 z  = *(const float4*)(row + j);
        float4 wa = *(const float4*)(Wa + j);
        float4 wb = *(const float4*)(Wb + j);
        da += z.x * wa.x + z.y * wa.y + z.z * wa.z + z.w * wa.w;
        db += z.x * wb.x + z.y * wb.y + z.z * wb.z + z.w * wb.w;
    }
    outA[t] = lrelu(da + ba[0]);
    outB[t] = lrelu(db + bb[0]);
}

// ---------------- tprob second layer: [N,64] @ [64,2] + bias, lrelu ----------------
__global__ void tprob_kernel(const float* __restrict__ P1, const float* __restrict__ Wp2,
                             const float* __restrict__ bp2, float* __restrict__ out, int n) {
    int i = blockIdx.x * blockDim.x + threadIdx.x;
    if (i >= n) return;
    const float* row = P1 + (size_t)i * HDIM;
    float d0 = 0.0f, d1 = 0.0f;
#pragma unroll
    for (int k = 0; k < HDIM; ++k) {
        float z = row[k];
        d0 += z * Wp2[k * 2 + 0];
        d1 += z * Wp2[k * 2 + 1];
    }
    out[i * 2 + 0] = lrelu(d0 + bp2[0]);
    out[i * 2 + 1] = lrelu(d1 + bp2[1]);
}

extern "C" void kernel_launch(void* const* d_in, const int* in_sizes, int n_in,
                              void* d_out, int out_size, void* d_ws, size_t ws_size,
                              hipStream_t stream) {
    const float* x     = (const float*)d_in[0];
    const int*   ei    = (const int*)d_in[1];
    const int*   treat = (const int*)d_in[2];
    const int*   ctrl  = (const int*)d_in[3];
    const float* W1  = (const float*)d_in[4];  const float* b1  = (const float*)d_in[5];
    const float* W2  = (const float*)d_in[6];  const float* b2  = (const float*)d_in[7];
    const float* Wy1 = (const float*)d_in[8];  const float* by1 = (const float*)d_in[9];
    const float* Wy0 = (const float*)d_in[10]; const float* by0 = (const float*)d_in[11];
    const float* Wp1 = (const float*)d_in[12]; const float* bp1 = (const float*)d_in[13];
    const float* Wp2 = (const float*)d_in[14]; const float* bp2 = (const float*)d_in[15];

    const int N = in_sizes[0] / INDIM;
    const int E = in_sizes[1] / 2;
    const int T = in_sizes[2];
    const int Cn = in_sizes[3];
    const int* e_src = ei;
    const int* e_dst = ei + E;

    // workspace: dinv [N] | buf0 [N*64] | buf1 [N*64]
    float* dinv = (float*)d_ws;
    float* buf0 = dinv + (((size_t)N + 63) & ~(size_t)63);
    float* buf1 = buf0 + (size_t)N * HDIM;

    // output slices (return order: y1, yc0, y0, yc1, tprob, xZ2)
    float* o_y1    = (float*)d_out;
    float* o_yc0   = o_y1 + T;
    float* o_y0    = o_yc0 + T;
    float* o_yc1   = o_y0 + Cn;
    float* o_tprob = o_yc1 + Cn;
    float* o_xZ2   = o_tprob + (size_t)2 * N;

    const int BLK = 256;
    auto cdiv = [](long long a, long long b) { return (int)((a + b - 1) / b); };
    const int gemm_blocks = cdiv(N / 16, 8);  // one 16x64 strip per wave, 8 waves/block
    const size_t smem128 = (size_t)INDIM * HDIM * sizeof(float);  // 32 KB
    const size_t smem64  = (size_t)HDIM * HDIM * sizeof(float);   // 16 KB

    // --- normalization: deg -> dinv ---
    deg_init_kernel<<<cdiv(N, BLK), BLK, 0, stream>>>(dinv, N);
    deg_count_kernel<<<cdiv(E, BLK), BLK, 0, stream>>>(e_dst, dinv, E);
    dinv_kernel<<<cdiv(N, BLK), BLK, 0, stream>>>(dinv, N);

    // --- conv1: H1 = x@W1 ; A1 = b1 + dinv^2*H1 + scatter ; Z1 = relu(A1) ---
    wmma_gemm_f32_kernel<<<gemm_blocks, BLK, smem128, stream>>>(x, W1, buf0, nullptr, N, INDIM, 0);
    conv_init_kernel<<<cdiv((long long)N * 16, BLK), BLK, 0, stream>>>(buf0, dinv, b1, buf1, N);
    conv_scatter_kernel<<<cdiv((long long)E * 16, BLK), BLK, 0, stream>>>(buf0, e_src, e_dst, dinv, buf1, E);
    relu_inplace_kernel<<<cdiv((long long)N * 16, BLK), BLK, 0, stream>>>(buf1, N * 16);

    // --- conv2: H2 = Z1@W2 ; xZ2 = b2 + dinv^2*H2 + scatter  (direct into d_out) ---
    wmma_gemm_f32_kernel<<<gemm_blocks, BLK, smem64, stream>>>(buf1, W2, buf0, nullptr, N, HDIM, 0);
    conv_init_kernel<<<cdiv((long long)N * 16, BLK), BLK, 0, stream>>>(buf0, dinv, b2, o_xZ2, N);
    conv_scatter_kernel<<<cdiv((long long)E * 16, BLK), BLK, 0, stream>>>(buf0, e_src, e_dst, dinv, o_xZ2, E);

    // --- outcome heads ---
    dual_head_kernel<<<cdiv(T, BLK), BLK, 0, stream>>>(o_xZ2, treat, T, Wy1, by1, o_y1, Wy0, by0, o_yc0);
    dual_head_kernel<<<cdiv(Cn, BLK), BLK, 0, stream>>>(o_xZ2, ctrl, Cn, Wy0, by0, o_y0, Wy1, by1, o_yc1);

    // --- propensity: P1 = lrelu(xZ2@Wp1 + bp1) via WMMA w/ fused epilogue, then [64,2] layer ---
    wmma_gemm_f32_kernel<<<gemm_blocks, BLK, smem64, stream>>>(o_xZ2, Wp1, buf0, bp1, N, HDIM, 1);
    tprob_kernel<<<cdiv(N, BLK), BLK, 0, stream>>>(buf0, Wp2, bp2, o_tprob, N);
}